// LCNSpiking_56229711839460
// MI455X (gfx1250) — compile-verified
//
#include <hip/hip_runtime.h>
#include <hip/hip_bf16.h>

typedef __attribute__((ext_vector_type(2))) float v2f;
typedef __attribute__((ext_vector_type(8))) float v8f;

#define LCN_K 25
#define NPB 8
#define BATCH 128
#define TSTEPS 16
#define IN_DIM 14400

#if defined(__has_builtin)
#if __has_builtin(__builtin_amdgcn_global_load_async_to_lds_b32)
#define HAVE_ASYNC_LDS 1
#endif
#endif

#ifdef HAVE_ASYNC_LDS
typedef __attribute__((address_space(1))) int g_int;   // global
typedef __attribute__((address_space(3))) int l_int;   // LDS
#endif

// ---------------------------------------------------------------------------
// Transpose x[:, t, :] (batch-major) -> xT[f][b] (feature-major, b contiguous)
// Async path: lanes read coalesced along f and scatter DIRECTLY to the
// transposed LDS position (per-lane LDS dest addr, stride-33 rows is
// bank-conflict free since gcd(33,64)=1), then s_wait_asynccnt + barrier.
// ---------------------------------------------------------------------------
__global__ void lcn_xt_transpose(const float* __restrict__ x,
                                 float* __restrict__ xT, int t) {
  __shared__ float tile[32][33];   // [f_local][b_local] (async) / [b][f] (sync)
  const int f0 = blockIdx.x * 32;
  const int b0 = blockIdx.y * 32;
#ifdef HAVE_ASYNC_LDS
  for (int j = threadIdx.y; j < 32; j += 8) {
    const float* gp = &x[((size_t)(b0 + j) * TSTEPS + t) * IN_DIM + f0 + threadIdx.x];
    __builtin_amdgcn_global_load_async_to_lds_b32(
        (g_int*)const_cast<float*>(gp),
        (l_int*)&tile[threadIdx.x][j], 0, 0);
  }
  asm volatile("s_wait_asynccnt 0x0" ::: "memory");
  __syncthreads();
  for (int j = threadIdx.y; j < 32; j += 8) {
    xT[(size_t)(f0 + j) * BATCH + b0 + threadIdx.x] = tile[j][threadIdx.x];
  }
#else
  for (int j = threadIdx.y; j < 32; j += 8) {
    tile[j][threadIdx.x] = __builtin_nontemporal_load(
        &x[((size_t)(b0 + j) * TSTEPS + t) * IN_DIM + f0 + threadIdx.x]);
  }
  __syncthreads();
  for (int j = threadIdx.y; j < 32; j += 8) {
    xT[(size_t)(f0 + j) * BATCH + b0 + threadIdx.x] = tile[threadIdx.x][j];
  }
#endif
}

// ---------------------------------------------------------------------------
// One LCN layer at one timestep: KNN gather (feature-major, fully coalesced
// 512B rows), 25-tap weighted sum, fused LIF (synaptic) update.
// threads = batch lanes (128), NPB neurons per block staged through LDS.
// ---------------------------------------------------------------------------
__global__ void lcn_layer_lif(const float* __restrict__ H,    // [nH][128]
                              const float* __restrict__ w,    // [d][25]
                              const float* __restrict__ bias, // [d]
                              const int*   __restrict__ knn,  // [d][25]
                              float* __restrict__ syn,        // [d][128]
                              float* __restrict__ mem,        // [d][128]
                              int d, int first) {
  __shared__ int   s_idx[NPB * LCN_K];
  __shared__ float s_w[NPB * LCN_K];
  __shared__ float s_b[NPB];
  const int tid = threadIdx.x;            // batch index 0..127
  const int d0  = blockIdx.x * NPB;
  const int nn  = (d - d0 < NPB) ? (d - d0) : NPB;
  const int cnt = nn * LCN_K;
  for (int j = tid; j < cnt; j += BATCH) {
    s_idx[j] = knn[(size_t)d0 * LCN_K + j];
    s_w[j]   = w[(size_t)d0 * LCN_K + j];
  }
  if (tid < nn) s_b[tid] = bias[d0 + tid];
  __syncthreads();

  for (int n = 0; n < nn; ++n) {
    // CDNA5 global_prefetch_b8 for the next neuron's gather rows
    if (n + 1 < nn && tid < LCN_K) {
      __builtin_prefetch(&H[(size_t)s_idx[(n + 1) * LCN_K + tid] * BATCH], 0, 3);
    }
    float acc = s_b[n];
#pragma unroll
    for (int k = 0; k < LCN_K; ++k) {
      acc = fmaf(H[(size_t)s_idx[n * LCN_K + k] * BATCH + tid],
                 s_w[n * LCN_K + k], acc);
    }
    const size_t off = (size_t)(d0 + n) * BATCH + tid;
    float s = 0.0f, m = 0.0f;
    if (!first) { s = syn[off]; m = mem[off]; }
    const float reset = (m > 1.0f) ? 1.0f : 0.0f;  // from PREVIOUS mem
    s = 0.9f * s + acc;
    m = 0.8f * m + s - reset;                       // THRESH == 1.0
    syn[off] = s;
    mem[off] = m;
  }
}

// ---------------------------------------------------------------------------
// Final FC: out[b, c] = sum_f mem4[f][b] * fcw[c][f] + fcb[c], c in {0,1}.
// fp32 WMMA: V_WMMA_F32_16X16X4_F32, one wave per 16-batch tile.
// fc_w staged in LDS zero-padded to 452 cols -> main loop is branch-free.
// A (16x4): lanes 0-15 hold K=0,1 in v0,v1; lanes 16-31 hold K=2,3.
// B (4x16): VGPR v, half h -> row K = v + 2*h, col N = lane%16.
// D (16x16): VGPR v, half h -> row M = v + 8*h, col N = lane%16.
// ---------------------------------------------------------------------------
#define FC_F 450
#define FC_FP 452
__global__ void lcn_fc_wmma(const float* __restrict__ H,    // [450][128]
                            const float* __restrict__ fcw,  // [2][450]
                            const float* __restrict__ fcb,  // [2]
                            float* __restrict__ out) {      // [128][2]
  __shared__ float s_w[2 * FC_FP];
  const int lane = threadIdx.x;     // 0..31, single wave
  for (int j = lane; j < 2 * FC_FP; j += 32) {
    const int c = j / FC_FP, f = j % FC_FP;
    s_w[j] = (f < FC_F) ? fcw[c * FC_F + f] : 0.0f;
  }
  __syncthreads();
  const int   half  = lane >> 4;
  const int   r     = lane & 15;
  const int   b     = blockIdx.x * 16 + r;
  const int   rc    = (r < 2) ? r : 0;            // clamp -> always in-bounds
  const float cmask = (r < 2) ? 1.0f : 0.0f;      // zero cols 2..15
  v8f acc = {};
  // Main loop: f in [0,448), every access provably in-bounds, zero guards.
  for (int f0 = 0; f0 < 448; f0 += 4) {
    const int fa = f0 + 2 * half;
    v2f a, bb;
    a.x  = H[(size_t)fa * BATCH + b];
    a.y  = H[(size_t)(fa + 1) * BATCH + b];
    bb.x = cmask * s_w[rc * FC_FP + fa];
    bb.y = cmask * s_w[rc * FC_FP + fa + 1];
    acc = __builtin_amdgcn_wmma_f32_16x16x4_f32(
        false, a, false, bb, (short)0, acc, false, false);
  }
  // Remainder f = 448..451: clamp addresses, mask values (no control flow).
  {
    const int   fa = 448 + 2 * half;              // 448 or 450
    const int   f0c = (fa     < FC_F) ? fa     : FC_F - 1;
    const int   f1c = (fa + 1 < FC_F) ? fa + 1 : FC_F - 1;
    const float m0 = (fa     < FC_F) ? 1.0f : 0.0f;
    const float m1 = (fa + 1 < FC_F) ? 1.0f : 0.0f;
    v2f a, bb;
    a.x  = m0 * H[(size_t)f0c * BATCH + b];
    a.y  = m1 * H[(size_t)f1c * BATCH + b];
    bb.x = cmask * s_w[rc * FC_FP + fa];          // rows 450/451 are padded 0
    bb.y = cmask * s_w[rc * FC_FP + fa + 1];
    acc = __builtin_amdgcn_wmma_f32_16x16x4_f32(
        false, a, false, bb, (short)0, acc, false, false);
  }
#pragma unroll
  for (int v = 0; v < 8; ++v) {
    const int m = v + half * 8;
    if (r < 2) out[(size_t)(blockIdx.x * 16 + m) * 2 + r] = acc[v] + fcb[r];
  }
}

// ---------------------------------------------------------------------------
// Host: 16 x (transpose + 5 layers) + 1 WMMA FC, all on `stream`.
// Scratch layout (fp32 elems): xT[14400*128], then syn_i/mem_i per layer.
// ---------------------------------------------------------------------------
extern "C" void kernel_launch(void* const* d_in, const int* in_sizes, int n_in,
                              void* d_out, int out_size, void* d_ws, size_t ws_size,
                              hipStream_t stream) {
  (void)in_sizes; (void)n_in; (void)out_size; (void)ws_size;
  static const int DIMS[5] = {7200, 3600, 1800, 900, 450};

  const float* x = (const float*)d_in[0];
  const float* w[5]; const float* bias[5]; const int* knn[5];
  for (int i = 0; i < 5; ++i) {
    w[i]    = (const float*)d_in[1 + 3 * i];
    bias[i] = (const float*)d_in[2 + 3 * i];
    knn[i]  = (const int*)  d_in[3 + 3 * i];
  }
  const float* fcw = (const float*)d_in[16];
  const float* fcb = (const float*)d_in[17];

  float* ws = (float*)d_ws;
  float* xT = ws;
  size_t off = (size_t)IN_DIM * BATCH;
  float* syn[5]; float* mem[5];
  for (int i = 0; i < 5; ++i) {
    syn[i] = ws + off; off += (size_t)DIMS[i] * BATCH;
    mem[i] = ws + off; off += (size_t)DIMS[i] * BATCH;
  }

  for (int t = 0; t < TSTEPS; ++t) {
    lcn_xt_transpose<<<dim3(IN_DIM / 32, BATCH / 32), dim3(32, 8), 0, stream>>>(
        x, xT, t);
    const float* H = xT;
    for (int i = 0; i < 5; ++i) {
      const int grid = (DIMS[i] + NPB - 1) / NPB;
      lcn_layer_lif<<<grid, BATCH, 0, stream>>>(
          H, w[i], bias[i], knn[i], syn[i], mem[i], DIMS[i], t == 0 ? 1 : 0);
      H = mem[i];
    }
  }
  lcn_fc_wmma<<<BATCH / 16, 32, 0, stream>>>(mem[4], fcw, fcb, (float*)d_out);
}